// PrRoIPool2D_1726576854721
// MI455X (gfx1250) — compile-verified
//
#include <hip/hip_runtime.h>

#define PHB 7
#define PWB 7
#define KC 6
#define FSCALE 0.0625f
#define FB 4
#define FC 64
#define FH 200
#define FW 304
#define RMAX 48          // padded patch extent (true max ~37)
#define NWAVE 8          // 256 threads

typedef __attribute__((ext_vector_type(2))) float v2f;
typedef __attribute__((ext_vector_type(8))) float v8f;

__device__ __forceinline__ float f_piece(float a, float la) {
    return la - 0.5f * la * la - a + 0.5f * a * a;
}

__global__ __launch_bounds__(256)
void prroi_wmma_kernel(const float* __restrict__ feat,
                       const float* __restrict__ rois,
                       float* __restrict__ out)
{
    // Weight matrices padded to 16 rows so every lane of a wave32 can load
    // its B/A operand unconditionally (rows 7..15 are zero).
    __shared__ float sWX[16][RMAX];             // dense x-weights (banded, zero-padded)
    __shared__ float sWY[16][RMAX];             // dense y-weights
    __shared__ float sT[NWAVE][RMAX][16];       // per-wave pass-1 result  T[ry][j]

    const int n   = blockIdx.x;
    const int tid = threadIdx.x;

    // ---- per-RoI scalars (uniform across the block) ----
    const float r0 = rois[n * 5 + 0];
    const float x1 = rois[n * 5 + 1] * FSCALE;
    const float y1 = rois[n * 5 + 2] * FSCALE;
    const float x2 = rois[n * 5 + 3] * FSCALE;
    const float y2 = rois[n * 5 + 4] * FSCALE;
    const int   bimg = (int)r0;
    const float binw = fmaxf(x2 - x1, 0.f) * (1.0f / PWB);
    const float binh = fmaxf(y2 - y1, 0.f) * (1.0f / PHB);
    const int   col0 = (int)floorf(x1);
    const int   row0 = (int)floorf(y1);

    int RX = (int)floorf(x1 + binw * (PWB - 1)) - col0 + KC + 1;
    int RY = (int)floorf(y1 + binh * (PHB - 1)) - row0 + KC + 1;
    RX = min(max(RX, 1), RMAX);
    RY = min(max(RY, 1), RMAX);
    const int KX  = (RX + 3) >> 2;    // K-steps (K=4) for x contraction, <=12
    const int KY  = (RY + 3) >> 2;    // K-steps for y contraction, <=12
    const int RYT = (RY + 15) >> 4;   // 16-row tiles in pass 1, <=3

    // ---- build dense weight matrices in LDS ----
    for (int t = tid; t < 16 * RMAX; t += 256) ((float*)sWX)[t] = 0.f;
    for (int t = tid; t < 16 * RMAX; t += 256) ((float*)sWY)[t] = 0.f;
    __syncthreads();

    if (tid < PWB + PHB) {
        const int   axis  = (tid >= PWB) ? 1 : 0;
        const int   j     = axis ? (tid - PWB) : tid;
        const float start = axis ? y1 : x1;
        const float bs    = axis ? binh : binw;
        const int   sz    = axis ? FH : FW;
        const int   o0    = axis ? row0 : col0;
        float*      Wrow  = axis ? sWY[j] : sWX[j];

        const float ws = start + bs * (float)j;
        const float we = ws + bs;
        const float s  = floorf(ws);
        float Wg[KC + 1];
        #pragma unroll
        for (int g = 0; g <= KC; ++g) Wg[g] = 0.f;
        #pragma unroll
        for (int t2 = 0; t2 < KC; ++t2) {
            const float cell = s + (float)t2;
            const bool  act  = cell < we;
            const float a0   = fmaxf(ws, cell);
            const float a1   = fminf(we, cell + 1.f);
            const float L    = act ? f_piece(a0 - cell, a1 - cell) : 0.f;
            const float R    = act ? f_piece(cell + 1.f - a1, cell + 1.f - a0) : 0.f;
            Wg[t2]     += L;
            Wg[t2 + 1] += R;
        }
        const int si = (int)s;
        #pragma unroll
        for (int g = 0; g <= KC; ++g) {
            const int gi = si + g;
            if (gi >= 0 && gi < sz) {
                const int r = gi - o0;
                if (r >= 0 && r < RMAX) Wrow[r] = Wg[g];
            }
        }
    }
    __syncthreads();

    // ---- WMMA lane mapping (wave32, 16x16x4 f32) ----
    const int wave = tid >> 5;
    const int lane = tid & 31;
    const int m15  = lane & 15;     // M (or N) index 0..15
    const int hh   = lane >> 4;     // selects K pair {2h, 2h+1}

    const float win = binw * binh;
    const float inv = (win > 0.f) ? (1.f / fmaxf(win, 1e-12f)) : 0.f;

    for (int ci = 0; ci < FC / NWAVE; ++ci) {
        const int c = wave * (FC / NWAVE) + ci;
        const float* fb = feat + ((size_t)bimg * FC + (size_t)c) * (size_t)(FH * FW);

        // -------- pass 1:  T[ry, j] = sum_rx  P[ry, rx] * WX[j, rx] --------
        for (int ryt = 0; ryt < RYT; ++ryt) {
            int row = row0 + ryt * 16 + m15;
            row = min(max(row, 0), FH - 1);
            const float* frow = fb + (size_t)row * FW;

            // 1-deep software pipeline on the global A operand
            const int rx0  = hh * 2;
            int cc0 = min(max(col0 + rx0,     0), FW - 1);
            int cc1 = min(max(col0 + rx0 + 1, 0), FW - 1);
            v2f a_cur;
            a_cur.x = frow[cc0];
            a_cur.y = frow[cc1];

            v8f acc = {};
            for (int ks = 0; ks < KX; ++ks) {
                const int rx = ks * 4 + hh * 2;
                v2f a_next = {};
                if (ks + 1 < KX) {                 // uniform condition
                    const int rxn = rx + 4;
                    const int n0  = min(max(col0 + rxn,     0), FW - 1);
                    const int n1  = min(max(col0 + rxn + 1, 0), FW - 1);
                    a_next.x = frow[n0];
                    a_next.y = frow[n1];
                }
                v2f b;                             // B(k=rx_local, n=j) = WX[j, rx]
                b.x = sWX[m15][rx];
                b.y = sWX[m15][rx + 1];
                acc = __builtin_amdgcn_wmma_f32_16x16x4_f32(
                        false, a_cur, false, b, (short)0, acc, false, false);
                a_cur = a_next;
            }
            // D(m,n): reg r -> m = r + 8*hh, n = m15  (cols 7..15 are exact zeros)
            #pragma unroll
            for (int r = 0; r < 8; ++r)
                sT[wave][ryt * 16 + r + hh * 8][m15] = acc[r];
        }

        // -------- pass 2:  out[i, j] = sum_ry  WY[i, ry] * T[ry, j] --------
        v8f acc2 = {};
        for (int ks = 0; ks < KY; ++ks) {
            const int ry = ks * 4 + hh * 2;
            v2f a;                           // A(m=i, k=ry) = WY[i, ry]
            a.x = sWY[m15][ry];
            a.y = sWY[m15][ry + 1];
            v2f b;                           // B(k=ry, n=j) = T[ry, j]
            b.x = sT[wave][ry][m15];
            b.y = sT[wave][ry + 1][m15];
            acc2 = __builtin_amdgcn_wmma_f32_16x16x4_f32(
                     false, a, false, b, (short)0, acc2, false, false);
        }

        // epilogue: lanes 0..6 (hh==0) hold i = r (0..6), n = j = m15
        if (hh == 0 && m15 < PWB) {
            float* op = out + (((size_t)n * FC + (size_t)c) * PHB) * PWB + m15;
            #pragma unroll
            for (int r = 0; r < PHB; ++r)
                op[r * PWB] = acc2[r] * inv;
        }
    }
}

extern "C" void kernel_launch(void* const* d_in, const int* in_sizes, int n_in,
                              void* d_out, int out_size, void* d_ws, size_t ws_size,
                              hipStream_t stream) {
    const float* feat = (const float*)d_in[0];
    const float* rois = (const float*)d_in[1];
    float* out = (float*)d_out;
    const int N = in_sizes[1] / 5;   // rois are [N,5]
    prroi_wmma_kernel<<<N, 256, 0, stream>>>(feat, rois, out);
}